// GATModel_47648367182527
// MI455X (gfx1250) — compile-verified
//
#include <hip/hip_runtime.h>
#include <math.h>
#include <stdint.h>

// ---------------------------------------------------------------------------
// GAT (3-layer) for MI455X / gfx1250.
//  - fp32 WMMA 16x16x4 GEMMs, B panel staged in LDS as paired float2 so each
//    B fragment is a single ds_load_b64 (8 waves share one 64KB panel).
//  - wave-per-edge gather/scatter attention aggregation (L2-resident).
// ---------------------------------------------------------------------------

typedef __attribute__((ext_vector_type(2))) float v2f;
typedef __attribute__((ext_vector_type(8))) float v8f;

#define NEG_SLOPE 0.2f
#define GEMM_WAVES 8   // waves (row tiles) per block

// ---- order-preserving float<->int map for atomicMax on floats --------------
__device__ __forceinline__ int f2ord(float f) {
    int i = __float_as_int(f);
    return (i >= 0) ? i : (i ^ 0x7fffffff);
}
__device__ __forceinline__ float ord2f(int i) {
    return __int_as_float((i >= 0) ? i : (i ^ 0x7fffffff));
}

__device__ __forceinline__ float waveReduceSum(float v) {
    for (int off = 16; off > 0; off >>= 1) v += __shfl_xor(v, off, 32);
    return v;
}

// ---------------------------------------------------------------------------
// GEMM: C[Nr,M] = A[Nr,K] * B[K,M], fp32, via V_WMMA_F32_16X16X4_F32.
// Block = 8 waves; each wave computes a 16x64 strip (4 accumulators).
// B panel (K x 64) staged in LDS as float2 {B[2k][c], B[2k+1][c]} so a
// fragment load is one ds_load_b64. K%4==0, M%64==0; Nr guarded per wave.
// A-frag (16x4 f32): lanes 0-15 row=l, K={0,1}; lanes 16-31 K={2,3}.
// B-frag (4x16 f32): vgpr0 lanes0-15 K=0 / lanes16-31 K=2; vgpr1 K=1 / K=3.
// C/D (16x16 f32): vgpr r -> rows r (lanes 0-15) and r+8 (lanes 16-31).
// ---------------------------------------------------------------------------
__global__ void __launch_bounds__(32 * GEMM_WAVES)
gemm_wmma_f32(const float* __restrict__ A, const float* __restrict__ B,
              float* __restrict__ C, int Nr, int K, int M) {
    __shared__ float2 Bs[(256 / 2) * 64];   // up to K=256 -> 64 KB

    const int tid  = threadIdx.x;
    const int wave = tid >> 5;
    const int lane = tid & 31;
    const int half = lane >> 4;     // 0 or 1
    const int l    = lane & 15;
    const int col0 = blockIdx.y * 64;
    const int row0 = (blockIdx.x * GEMM_WAVES + wave) * 16;

    // cooperative fill of the shared B panel: Bs[kk*64+c] = {B[2kk][c0+c], B[2kk+1][c0+c]}
    const int panel = (K >> 1) * 64;
    for (int idx = tid; idx < panel; idx += 32 * GEMM_WAVES) {
        int kk = idx >> 6;
        int c  = idx & 63;
        float2 v;
        v.x = B[(size_t)(2 * kk)     * M + col0 + c];
        v.y = B[(size_t)(2 * kk + 1) * M + col0 + c];
        Bs[idx] = v;
    }
    __syncthreads();

    if (row0 >= Nr) return;

    v8f acc0 = {}, acc1 = {}, acc2 = {}, acc3 = {};
    const float* Arow = A + (size_t)(row0 + l) * K + 2 * half;
    for (int k = 0; k < K; k += 4) {
        v2f a;
        a.x = Arow[k];
        a.y = Arow[k + 1];
        const float2* Bp = &Bs[((k >> 1) + half) * 64 + l];
        float2 f0 = Bp[0], f1 = Bp[16], f2 = Bp[32], f3 = Bp[48];
        v2f b0, b1, b2, b3;
        b0.x = f0.x; b0.y = f0.y;
        b1.x = f1.x; b1.y = f1.y;
        b2.x = f2.x; b2.y = f2.y;
        b3.x = f3.x; b3.y = f3.y;
        acc0 = __builtin_amdgcn_wmma_f32_16x16x4_f32(false, a, false, b0, (short)0, acc0, false, false);
        acc1 = __builtin_amdgcn_wmma_f32_16x16x4_f32(false, a, false, b1, (short)0, acc1, false, false);
        acc2 = __builtin_amdgcn_wmma_f32_16x16x4_f32(false, a, false, b2, (short)0, acc2, false, false);
        acc3 = __builtin_amdgcn_wmma_f32_16x16x4_f32(false, a, false, b3, (short)0, acc3, false, false);
    }

    const int crow = row0 + half * 8;
    float* Cp = C + (size_t)crow * M + col0 + l;
    #pragma unroll
    for (int r = 0; r < 8; ++r) {
        Cp[(size_t)r * M + 0]  = acc0[r];
        Cp[(size_t)r * M + 16] = acc1[r];
        Cp[(size_t)r * M + 32] = acc2[r];
        Cp[(size_t)r * M + 48] = acc3[r];
    }
}

// ---------------------------------------------------------------------------
// Per-node attention logits: al_src[n,h] = sum_c h[n,h,c]*a_src[h,c] (dst same)
// ---------------------------------------------------------------------------
template <int H>
__global__ void attn_logits(const float* __restrict__ hfeat,
                            const float* __restrict__ a_src,
                            const float* __restrict__ a_dst,
                            float* __restrict__ alsrc, float* __restrict__ aldst,
                            int Nn, int C) {
    int i = blockIdx.x * blockDim.x + threadIdx.x;
    if (i >= Nn * H) return;
    int n = i / H, hd = i % H;
    const float* hp = hfeat + (size_t)n * H * C + (size_t)hd * C;
    const float* as = a_src + hd * C;
    const float* ad = a_dst + hd * C;
    float ss = 0.f, sd = 0.f;
    for (int c = 0; c < C; ++c) {
        float v = hp[c];
        ss += v * as[c];
        sd += v * ad[c];
    }
    alsrc[i] = ss;
    aldst[i] = sd;
}

// ---------------------------------------------------------------------------
// Init segment buffers: m = ord(-FLT_MAX), s = 0, agg = 0
// ---------------------------------------------------------------------------
__global__ void seg_init(int* __restrict__ mord, float* __restrict__ ssum,
                         float* __restrict__ agg, int NH, int NF) {
    int i = blockIdx.x * blockDim.x + threadIdx.x;
    if (i < NH) {
        mord[i] = f2ord(-3.402823466e38f);
        ssum[i] = 0.f;
    }
    if (i < NF) agg[i] = 0.f;
}

// ---------------------------------------------------------------------------
// Pass 1: logit = leaky_relu(al_src[src]+al_dst[dst]); segment max via atomic
// Edge e>=E is the self loop (src=dst=e-E).
// ---------------------------------------------------------------------------
template <int H>
__global__ void edge_logit_max(const int* __restrict__ esrc, const int* __restrict__ edst,
                               const float* __restrict__ alsrc, const float* __restrict__ aldst,
                               float* __restrict__ elog, int* __restrict__ mord,
                               int E, int Nn) {
    int i = blockIdx.x * blockDim.x + threadIdx.x;
    int ET = E + Nn;
    if (i >= ET * H) return;
    int e = i / H, h = i % H;
    int s, d;
    if (e < E) { s = esrc[e]; d = edst[e]; } else { s = d = e - E; }
    float v = alsrc[s * H + h] + aldst[d * H + h];
    v = (v > 0.f) ? v : NEG_SLOPE * v;
    elog[i] = v;
    atomicMax(&mord[d * H + h], f2ord(v));
}

// ---------------------------------------------------------------------------
// Pass 2: e = exp(logit - m[dst]); segment sum
// ---------------------------------------------------------------------------
template <int H>
__global__ void edge_exp_sum(const int* __restrict__ edst,
                             const int* __restrict__ mord, float* __restrict__ elog,
                             float* __restrict__ ssum, int E, int Nn) {
    int i = blockIdx.x * blockDim.x + threadIdx.x;
    int ET = E + Nn;
    if (i >= ET * H) return;
    int e = i / H, h = i % H;
    int d = (e < E) ? edst[e] : (e - E);
    float m = ord2f(mord[d * H + h]);
    if (m < -1e37f) m = 0.f;   // isfinite guard (ref: where(isfinite(m), m, 0))
    float ev = expf(elog[i] - m);
    elog[i] = ev;
    atomicAdd(&ssum[d * H + h], ev);
}

// ---------------------------------------------------------------------------
// Pass 3: agg[dst] += (e/(s[dst]+eps)) * h[src]   -- one wave per edge,
// 256-float feature row striped across 32 lanes; per-head alpha hoisted.
// ---------------------------------------------------------------------------
template <int H, int C>
__global__ void edge_aggregate(const int* __restrict__ esrc, const int* __restrict__ edst,
                               const float* __restrict__ elog, const float* __restrict__ ssum,
                               const float* __restrict__ hfeat, float* __restrict__ agg,
                               int E, int Nn) {
    int wave = (blockIdx.x * blockDim.x + threadIdx.x) >> 5;
    int lane = threadIdx.x & 31;
    int ET = E + Nn;
    if (wave >= ET) return;
    int s, d;
    if (wave < E) { s = esrc[wave]; d = edst[wave]; } else { s = d = wave - E; }
    float alpha[H];
    #pragma unroll
    for (int h = 0; h < H; ++h)
        alpha[h] = elog[(size_t)wave * H + h] / (ssum[d * H + h] + 1e-16f);
    const int F = H * C;
    const float* hs = hfeat + (size_t)s * F;
    float* ap = agg + (size_t)d * F;
    #pragma unroll
    for (int c = lane; c < F; c += 32)
        atomicAdd(&ap[c], alpha[c / C] * hs[c]);
}

// scalar variant for layer 3 (H=1, C=1)
__global__ void edge_aggregate1(const int* __restrict__ esrc, const int* __restrict__ edst,
                                const float* __restrict__ elog, const float* __restrict__ ssum,
                                const float* __restrict__ hfeat, float* __restrict__ agg,
                                int E, int Nn) {
    int e = blockIdx.x * blockDim.x + threadIdx.x;
    int ET = E + Nn;
    if (e >= ET) return;
    int s, d;
    if (e < E) { s = esrc[e]; d = edst[e]; } else { s = d = e - E; }
    float alpha = elog[e] / (ssum[d] + 1e-16f);
    atomicAdd(&agg[d], alpha * hfeat[s]);
}

// ---------------------------------------------------------------------------
// out = elu(v + bias[c mod F]) in place
// ---------------------------------------------------------------------------
__global__ void bias_elu(float* __restrict__ v, const float* __restrict__ b,
                         int Nn, int F) {
    int i = blockIdx.x * blockDim.x + threadIdx.x;
    if (i >= Nn * F) return;
    float x = v[i] + b[i & (F - 1)];   // F = 256 (power of two)
    v[i] = (x > 0.f) ? x : (expf(x) - 1.f);
}

// h3[n] = h[n,:] . W3[:,0]  (wave-per-node shuffle reduction)
__global__ void dot_w3(const float* __restrict__ X, const float* __restrict__ W,
                       float* __restrict__ y, int Nn, int K) {
    int wave = (blockIdx.x * blockDim.x + threadIdx.x) >> 5;
    int lane = threadIdx.x & 31;
    if (wave >= Nn) return;
    const float* xp = X + (size_t)wave * K;
    float s = 0.f;
    for (int c = lane; c < K; c += 32) s += xp[c] * W[c];
    s = waveReduceSum(s);
    if (lane == 0) y[wave] = s;
}

__global__ void attn3(const float* __restrict__ h3, const float* __restrict__ as,
                      const float* __restrict__ ad, float* __restrict__ alsrc,
                      float* __restrict__ aldst, int Nn) {
    int i = blockIdx.x * blockDim.x + threadIdx.x;
    if (i >= Nn) return;
    float v = h3[i];
    alsrc[i] = v * as[0];
    aldst[i] = v * ad[0];
}

__global__ void add_bias_out(const float* __restrict__ agg, const float* __restrict__ b,
                             float* __restrict__ out, int Nn) {
    int i = blockIdx.x * blockDim.x + threadIdx.x;
    if (i >= Nn) return;
    out[i] = agg[i] + b[0];   // concat=False, H=1 -> mean over heads is identity
}

// ---------------------------------------------------------------------------
static inline char* carve(char*& p, size_t bytes) {
    char* r = p;
    p += (bytes + 255) & ~(size_t)255;
    return r;
}
static inline int cdiv(int a, int b) { return (a + b - 1) / b; }

extern "C" void kernel_launch(void* const* d_in, const int* in_sizes, int n_in,
                              void* d_out, int out_size, void* d_ws, size_t ws_size,
                              hipStream_t stream) {
    (void)n_in; (void)out_size; (void)ws_size;
    const int IN = 128, HID = 64, H = 4, F = H * HID;     // F = 256
    const int Nn = in_sizes[0] / IN;                      // 50000
    const int E  = in_sizes[1] / 2;                       // 650000
    const int ET = E + Nn;                                // 700000

    const float* x   = (const float*)d_in[0];
    const int* esrc  = (const int*)d_in[1];
    const int* edst  = esrc + E;
    const float* W1  = (const float*)d_in[2];
    const float* as1 = (const float*)d_in[3];
    const float* ad1 = (const float*)d_in[4];
    const float* b1  = (const float*)d_in[5];
    const float* W2  = (const float*)d_in[6];
    const float* as2 = (const float*)d_in[7];
    const float* ad2 = (const float*)d_in[8];
    const float* b2  = (const float*)d_in[9];
    const float* W3  = (const float*)d_in[10];
    const float* as3 = (const float*)d_in[11];
    const float* ad3 = (const float*)d_in[12];
    const float* b3  = (const float*)d_in[13];

    char* p = (char*)d_ws;
    float* bufA  = (float*)carve(p, (size_t)Nn * F * sizeof(float));  // GEMM out
    float* bufB  = (float*)carve(p, (size_t)Nn * F * sizeof(float));  // aggregation
    float* alsrc = (float*)carve(p, (size_t)Nn * H * sizeof(float));
    float* aldst = (float*)carve(p, (size_t)Nn * H * sizeof(float));
    int*   mord  = (int*)  carve(p, (size_t)Nn * H * sizeof(int));
    float* ssum  = (float*)carve(p, (size_t)Nn * H * sizeof(float));
    float* elog  = (float*)carve(p, (size_t)ET * H * sizeof(float));
    float* h3    = (float*)carve(p, (size_t)Nn * sizeof(float));
    float* agg3  = (float*)carve(p, (size_t)Nn * sizeof(float));

    const int B256 = 256;
    const int rowTiles = Nn / 16;                               // 3125
    dim3 gGemm(cdiv(rowTiles, GEMM_WAVES), F / 64);
    const int gemmThreads = 32 * GEMM_WAVES;

    // ================= Layer 1 =================
    gemm_wmma_f32<<<gGemm, gemmThreads, 0, stream>>>(x, W1, bufA, Nn, IN, F);
    attn_logits<4><<<cdiv(Nn * H, B256), B256, 0, stream>>>(bufA, as1, ad1, alsrc, aldst, Nn, HID);
    seg_init<<<cdiv(Nn * F, B256), B256, 0, stream>>>(mord, ssum, bufB, Nn * H, Nn * F);
    edge_logit_max<4><<<cdiv(ET * H, B256), B256, 0, stream>>>(esrc, edst, alsrc, aldst, elog, mord, E, Nn);
    edge_exp_sum<4><<<cdiv(ET * H, B256), B256, 0, stream>>>(edst, mord, elog, ssum, E, Nn);
    edge_aggregate<4, 64><<<cdiv(ET * 32, B256), B256, 0, stream>>>(esrc, edst, elog, ssum, bufA, bufB, E, Nn);
    bias_elu<<<cdiv(Nn * F, B256), B256, 0, stream>>>(bufB, b1, Nn, F);

    // ================= Layer 2 =================
    gemm_wmma_f32<<<gGemm, gemmThreads, 0, stream>>>(bufB, W2, bufA, Nn, F, F);
    attn_logits<4><<<cdiv(Nn * H, B256), B256, 0, stream>>>(bufA, as2, ad2, alsrc, aldst, Nn, HID);
    seg_init<<<cdiv(Nn * F, B256), B256, 0, stream>>>(mord, ssum, bufB, Nn * H, Nn * F);
    edge_logit_max<4><<<cdiv(ET * H, B256), B256, 0, stream>>>(esrc, edst, alsrc, aldst, elog, mord, E, Nn);
    edge_exp_sum<4><<<cdiv(ET * H, B256), B256, 0, stream>>>(edst, mord, elog, ssum, E, Nn);
    edge_aggregate<4, 64><<<cdiv(ET * 32, B256), B256, 0, stream>>>(esrc, edst, elog, ssum, bufA, bufB, E, Nn);
    bias_elu<<<cdiv(Nn * F, B256), B256, 0, stream>>>(bufB, b2, Nn, F);

    // ================= Layer 3 (H=1, C=1) =================
    dot_w3<<<cdiv(Nn * 32, B256), B256, 0, stream>>>(bufB, W3, h3, Nn, F);
    attn3<<<cdiv(Nn, B256), B256, 0, stream>>>(h3, as3, ad3, alsrc, aldst, Nn);
    seg_init<<<cdiv(Nn, B256), B256, 0, stream>>>(mord, ssum, agg3, Nn, Nn);
    edge_logit_max<1><<<cdiv(ET, B256), B256, 0, stream>>>(esrc, edst, alsrc, aldst, elog, mord, E, Nn);
    edge_exp_sum<1><<<cdiv(ET, B256), B256, 0, stream>>>(edst, mord, elog, ssum, E, Nn);
    edge_aggregate1<<<cdiv(ET, B256), B256, 0, stream>>>(esrc, edst, elog, ssum, h3, agg3, E, Nn);
    add_bias_out<<<cdiv(Nn, B256), B256, 0, stream>>>(agg3, b3, (float*)d_out, Nn);
}